// VoxelSetAbstraction_87454124081814
// MI455X (gfx1250) — compile-verified
//
#include <hip/hip_runtime.h>

typedef __attribute__((ext_vector_type(2))) float v2f;
typedef __attribute__((ext_vector_type(8))) float v8f;

static constexpr float R2(double r) { return (float)(r * r); }

// ---------------------------------------------------------------------------
// BEV bilinear interpolation: one block per (b,keypoint), 128 threads = channels
// ---------------------------------------------------------------------------
__global__ void bev_interp(const float* __restrict__ bev, const float* __restrict__ kp,
                           float* __restrict__ pf, int M) {
  const int Hh = 100, Ww = 126, C = 128;
  int bm = blockIdx.x;
  int c  = threadIdx.x;
  int b  = bm / M;
  float x = kp[(size_t)bm * 3 + 0];
  float y = kp[(size_t)bm * 3 + 1];
  float kx = (x - (-50.4f)) / 0.1f / 8.0f;
  float ky = (y - (-40.0f)) / 0.1f / 8.0f;
  int xf = (int)floorf(kx), yf = (int)floorf(ky);
  int x0 = min(max(xf, 0), Ww - 1);
  int x1 = min(max(xf + 1, 0), Ww - 1);
  int y0 = min(max(yf, 0), Hh - 1);
  int y1 = min(max(yf + 1, 0), Hh - 1);
  float x0f = (float)x0, x1f = (float)x1, y0f = (float)y0, y1f = (float)y1;
  const float* bb = bev + ((size_t)b * C + c) * Hh * Ww;
  float Ia = bb[y0 * Ww + x0], Ib = bb[y1 * Ww + x0];
  float Ic = bb[y0 * Ww + x1], Id = bb[y1 * Ww + x1];
  float wa = (x1f - kx) * (y1f - ky);
  float wb = (x1f - kx) * (ky - y0f);
  float wc = (kx - x0f) * (y1f - ky);
  float wd = (kx - x0f) * (ky - y0f);
  pf[(size_t)bm * 512 + c] = Ia * wa + Ib * wb + Ic * wc + Id * wd;
}

// ---------------------------------------------------------------------------
// Dual-radius ball query: one wave32 per (b,keypoint). Ascending-index
// selection via ballot+prefix compaction == reference's sort-of-candidates.
// Pads remaining slots with first valid index (N-1 if empty), writes empty flags.
// ---------------------------------------------------------------------------
__global__ void group_select(const float* __restrict__ xyz, int N,
                             const float* __restrict__ kp, int M, int BM,
                             float r0sq, float r1sq, int ns0, int ns1, int stride,
                             int* __restrict__ idxb, int* __restrict__ emp) {
  int wave = (blockIdx.x * blockDim.x + threadIdx.x) >> 5;
  int lane = threadIdx.x & 31;
  if (wave >= BM) return;
  int bm = wave;
  int b = bm / M;
  const float* xz = xyz + (size_t)b * N * 3;
  float kx = kp[(size_t)bm * 3 + 0];
  float ky = kp[(size_t)bm * 3 + 1];
  float kz = kp[(size_t)bm * 3 + 2];
  int* out = idxb + (size_t)bm * stride;
  int cnt0 = 0, cnt1 = 0;
  int first0 = N - 1, first1 = N - 1;
  unsigned lm = (1u << lane) - 1u;
  for (int base = 0; base < N; base += 32) {
    int i = base + lane;
    float d2 = 3.0e38f;
    if (i < N) {
      float dx = kx - xz[(size_t)i * 3 + 0];
      float dy = ky - xz[(size_t)i * 3 + 1];
      float dz = kz - xz[(size_t)i * 3 + 2];
      d2 = dx * dx + dy * dy + dz * dz;
    }
    bool h0 = d2 < r0sq;
    bool h1 = d2 < r1sq;
    unsigned m0 = (unsigned)__ballot(h0);
    unsigned m1 = (unsigned)__ballot(h1);
    if (m0) {
      if (cnt0 == 0) first0 = __shfl(i, __ffs(m0) - 1, 32);
      if (h0) {
        int p = cnt0 + __popc(m0 & lm);
        if (p < ns0) out[p] = i;
      }
      cnt0 += __popc(m0);
    }
    if (m1) {
      if (cnt1 == 0) first1 = __shfl(i, __ffs(m1) - 1, 32);
      if (h1) {
        int p = cnt1 + __popc(m1 & lm);
        if (p < ns1) out[ns0 + p] = i;
      }
      cnt1 += __popc(m1);
    }
    if (cnt0 >= ns0 && cnt1 >= ns1) break;
  }
  for (int p = cnt0 + lane; p < ns0; p += 32) out[p] = first0;
  for (int p = cnt1 + lane; p < ns1; p += 32) out[ns0 + p] = first1;
  if (lane == 0) {
    emp[(size_t)bm * 2 + 0] = (cnt0 == 0) ? 1 : 0;
    emp[(size_t)bm * 2 + 1] = (cnt1 == 0) ? 1 : 0;
  }
}

// ---------------------------------------------------------------------------
// Grouped 2-layer shared MLP + maxpool, fp32 WMMA (V_WMMA_F32_16X16X4_F32).
// Block = (H/16) waves, one block per (b,keypoint). NS in {16,32}.
// ---------------------------------------------------------------------------
template <int CIN, int H, int NS>
__global__ void sa_mlp(const float* __restrict__ xyz, const float* __restrict__ feats,
                       int N, int F, const float* __restrict__ kp, int M,
                       const int* __restrict__ idxb, int idx_stride, int idx_off,
                       const int* __restrict__ emp, int emp_off,
                       const float* __restrict__ W1, const float* __restrict__ s1,
                       const float* __restrict__ b1, const float* __restrict__ W2,
                       const float* __restrict__ s2, const float* __restrict__ b2,
                       float* __restrict__ pf, int pf_off) {
  constexpr int NW = H / 16;
  constexpr int RT = NS / 16;
  constexpr int CINP = (CIN + 3) & ~3;
  __shared__ float g[NS][CINP];
  __shared__ float hid[NS][H];
  int bm = blockIdx.x;
  int b = bm / M;
  int tid = threadIdx.x;
  const int* idx = idxb + (size_t)bm * idx_stride + idx_off;
  const float* xz = xyz + (size_t)b * N * 3;
  const float* ft = feats ? (feats + (size_t)b * N * F) : nullptr;
  float kx = kp[(size_t)bm * 3 + 0];
  float ky = kp[(size_t)bm * 3 + 1];
  float kz = kp[(size_t)bm * 3 + 2];

  // gather grouped features into LDS (zero-padded K columns)
  for (int e = tid; e < NS * CINP; e += NW * 32) {
    int s = e / CINP, c = e % CINP;
    int id = idx[s];
    float v = 0.0f;
    if (c < 3) {
      float p = xz[(size_t)id * 3 + c];
      v = p - (c == 0 ? kx : (c == 1 ? ky : kz));
    } else if (c < CIN) {
      v = ft[(size_t)id * F + (c - 3)];
    }
    g[s][c] = v;
  }
  __syncthreads();

  int w = tid >> 5;
  int lane = tid & 31;
  int hc = lane & 15;
  bool hilane = lane >= 16;
  int col = w * 16 + hc;
  float sc1 = s1[col], bi1 = b1[col];
  float sc2 = s2[col], bi2 = b2[col];

  // layer 1
  for (int rt = 0; rt < RT; ++rt) {
    v8f acc = {0.f, 0.f, 0.f, 0.f, 0.f, 0.f, 0.f, 0.f};
    int r = rt * 16 + hc;
    for (int k0 = 0; k0 < CINP; k0 += 4) {
      int ka = k0 + (hilane ? 2 : 0);
      v2f a, bf;
      a.x = g[r][ka];
      a.y = g[r][ka + 1];
      bf.x = (ka < CIN) ? W1[(size_t)ka * H + col] : 0.0f;
      bf.y = (ka + 1 < CIN) ? W1[(size_t)(ka + 1) * H + col] : 0.0f;
      acc = __builtin_amdgcn_wmma_f32_16x16x4_f32(false, a, false, bf, (short)0, acc,
                                                  false, false);
    }
#pragma unroll
    for (int v = 0; v < 8; ++v) {
      int row = rt * 16 + v + (hilane ? 8 : 0);
      float h = acc[v] * sc1 + bi1;
      hid[row][col] = h > 0.0f ? h : 0.0f;
    }
  }
  __syncthreads();

  // layer 2 + maxpool over samples
  float cm = -3.0e38f;
  for (int rt = 0; rt < RT; ++rt) {
    v8f acc = {0.f, 0.f, 0.f, 0.f, 0.f, 0.f, 0.f, 0.f};
    int r = rt * 16 + hc;
    for (int k0 = 0; k0 < H; k0 += 4) {
      int ka = k0 + (hilane ? 2 : 0);
      v2f a, bf;
      a.x = hid[r][ka];
      a.y = hid[r][ka + 1];
      bf.x = W2[(size_t)ka * H + col];
      bf.y = W2[(size_t)(ka + 1) * H + col];
      acc = __builtin_amdgcn_wmma_f32_16x16x4_f32(false, a, false, bf, (short)0, acc,
                                                  false, false);
    }
#pragma unroll
    for (int v = 0; v < 8; ++v) {
      float y = acc[v] * sc2 + bi2;
      y = y > 0.0f ? y : 0.0f;
      cm = fmaxf(cm, y);
    }
  }
  cm = fmaxf(cm, __shfl_xor(cm, 16, 32));
  if (!hilane) {
    int e = emp[(size_t)bm * 2 + emp_off];
    pf[(size_t)bm * 512 + pf_off + col] = e ? 0.0f : cm;
  }
}

// ---------------------------------------------------------------------------
// Fusion GEMM: (BM x 512) x (512 x 32), scale+bias+ReLU. 16-row tile per block,
// 2 waves (16 cols each), K=512 in 128 fp32 WMMA steps, A tile staged in LDS.
// ---------------------------------------------------------------------------
__global__ void fusion_gemm(const float* __restrict__ pf, const float* __restrict__ Wf,
                            const float* __restrict__ sc, const float* __restrict__ bi,
                            float* __restrict__ out) {
  __shared__ float A[16][512];
  int rb = blockIdx.x * 16;
  int tid = threadIdx.x;  // 64
  for (int e = tid; e < 16 * 512; e += 64) {
    int r = e >> 9, k = e & 511;
    A[r][k] = pf[(size_t)(rb + r) * 512 + k];
  }
  __syncthreads();
  int w = tid >> 5;
  int lane = tid & 31;
  int hc = lane & 15;
  bool hilane = lane >= 16;
  int col = w * 16 + hc;
  v8f acc = {0.f, 0.f, 0.f, 0.f, 0.f, 0.f, 0.f, 0.f};
  for (int k0 = 0; k0 < 512; k0 += 4) {
    int ka = k0 + (hilane ? 2 : 0);
    v2f a, bf;
    a.x = A[hc][ka];
    a.y = A[hc][ka + 1];
    bf.x = Wf[(size_t)ka * 32 + col];
    bf.y = Wf[(size_t)(ka + 1) * 32 + col];
    acc = __builtin_amdgcn_wmma_f32_16x16x4_f32(false, a, false, bf, (short)0, acc,
                                                false, false);
  }
  float s = sc[col], bb = bi[col];
#pragma unroll
  for (int v = 0; v < 8; ++v) {
    float y = acc[v] * s + bb;
    y = y > 0.0f ? y : 0.0f;
    out[(size_t)(rb + v + (hilane ? 8 : 0)) * 32 + col] = y;
  }
}

// ---------------------------------------------------------------------------
extern "C" void kernel_launch(void* const* d_in, const int* in_sizes, int n_in,
                              void* d_out, int out_size, void* d_ws, size_t ws_size,
                              hipStream_t stream) {
  (void)in_sizes; (void)n_in; (void)out_size; (void)ws_size;
  const int B = 2, M = 2048, BM = B * M;
  auto P = [&](int i) { return (const float*)d_in[i]; };

  const float* points = P(0);
  const float* keyp   = P(1);
  const float* bev    = P(2);
  const float* xyzs[4] = {P(3), P(5), P(7), P(9)};
  const float* ftss[4] = {P(4), P(6), P(8), P(10)};
  const int Ns[4] = {16000, 8000, 4000, 2000};
  const int Fs[4] = {16, 32, 64, 64};

  float* pf  = (float*)d_ws;                                          // BM x 512
  int* idxb  = (int*)((char*)d_ws + (size_t)BM * 512 * sizeof(float)); // BM x 48
  int* emp   = idxb + (size_t)BM * 48;                                 // BM x 2
  float* out = (float*)d_out;

  // [0:128) BEV
  bev_interp<<<BM, 128, 0, stream>>>(bev, keyp, pf, M);

  // raw points: radii (0.4,0.8), ns (16,16), CIN=3, H=16 -> pf [128:160)
  group_select<<<BM / 8, 256, 0, stream>>>(points, 20000, keyp, M, BM,
                                           R2(0.4), R2(0.8), 16, 16, 32, idxb, emp);
  sa_mlp<3, 16, 16><<<BM, 32, 0, stream>>>(points, nullptr, 20000, 0, keyp, M,
      idxb, 32, 0, emp, 0, P(11), P(12), P(13), P(14), P(15), P(16), pf, 128);
  sa_mlp<3, 16, 16><<<BM, 32, 0, stream>>>(points, nullptr, 20000, 0, keyp, M,
      idxb, 32, 16, emp, 1, P(17), P(18), P(19), P(20), P(21), P(22), pf, 144);

  // src1: radii (0.4,0.8), ns (16,16), CIN=19, H=16 -> pf [160:192)
  group_select<<<BM / 8, 256, 0, stream>>>(xyzs[0], Ns[0], keyp, M, BM,
                                           R2(0.4), R2(0.8), 16, 16, 32, idxb, emp);
  sa_mlp<19, 16, 16><<<BM, 32, 0, stream>>>(xyzs[0], ftss[0], Ns[0], Fs[0], keyp, M,
      idxb, 32, 0, emp, 0, P(23), P(24), P(25), P(26), P(27), P(28), pf, 160);
  sa_mlp<19, 16, 16><<<BM, 32, 0, stream>>>(xyzs[0], ftss[0], Ns[0], Fs[0], keyp, M,
      idxb, 32, 16, emp, 1, P(29), P(30), P(31), P(32), P(33), P(34), pf, 176);

  // src2: radii (0.8,1.2), ns (16,32), CIN=35, H=32 -> pf [192:256)
  group_select<<<BM / 8, 256, 0, stream>>>(xyzs[1], Ns[1], keyp, M, BM,
                                           R2(0.8), R2(1.2), 16, 32, 48, idxb, emp);
  sa_mlp<35, 32, 16><<<BM, 64, 0, stream>>>(xyzs[1], ftss[1], Ns[1], Fs[1], keyp, M,
      idxb, 48, 0, emp, 0, P(35), P(36), P(37), P(38), P(39), P(40), pf, 192);
  sa_mlp<35, 32, 32><<<BM, 64, 0, stream>>>(xyzs[1], ftss[1], Ns[1], Fs[1], keyp, M,
      idxb, 48, 16, emp, 1, P(41), P(42), P(43), P(44), P(45), P(46), pf, 224);

  // src3: radii (1.2,2.4), ns (16,32), CIN=67, H=64 -> pf [256:384)
  group_select<<<BM / 8, 256, 0, stream>>>(xyzs[2], Ns[2], keyp, M, BM,
                                           R2(1.2), R2(2.4), 16, 32, 48, idxb, emp);
  sa_mlp<67, 64, 16><<<BM, 128, 0, stream>>>(xyzs[2], ftss[2], Ns[2], Fs[2], keyp, M,
      idxb, 48, 0, emp, 0, P(47), P(48), P(49), P(50), P(51), P(52), pf, 256);
  sa_mlp<67, 64, 32><<<BM, 128, 0, stream>>>(xyzs[2], ftss[2], Ns[2], Fs[2], keyp, M,
      idxb, 48, 16, emp, 1, P(53), P(54), P(55), P(56), P(57), P(58), pf, 320);

  // src4: radii (2.4,4.8), ns (16,32), CIN=67, H=64 -> pf [384:512)
  group_select<<<BM / 8, 256, 0, stream>>>(xyzs[3], Ns[3], keyp, M, BM,
                                           R2(2.4), R2(4.8), 16, 32, 48, idxb, emp);
  sa_mlp<67, 64, 16><<<BM, 128, 0, stream>>>(xyzs[3], ftss[3], Ns[3], Fs[3], keyp, M,
      idxb, 48, 0, emp, 0, P(59), P(60), P(61), P(62), P(63), P(64), pf, 384);
  sa_mlp<67, 64, 32><<<BM, 128, 0, stream>>>(xyzs[3], ftss[3], Ns[3], Fs[3], keyp, M,
      idxb, 48, 16, emp, 1, P(65), P(66), P(67), P(68), P(69), P(70), pf, 448);

  // fusion: (BM x 512) @ (512 x 32) + scale/bias + ReLU
  fusion_gemm<<<BM / 16, 64, 0, stream>>>(pf, P(71), P(72), P(73), out);
}